// LPKTNet_88656714925283
// MI455X (gfx1250) — compile-verified
//
#include <hip/hip_runtime.h>
#include <hip/hip_bf16.h>

// ---------------------------------------------------------------------------
// LPKT scan for MI455X (gfx1250, wave32, WGP with 320KB LDS)
//
// K1: le[b,s]      = b1 + W1 @ [ex|at|ans]                 (batched matvec)
// K2: preL/preG    = b + W[:, :384] @ [learn_pre|itv|lv]   (non-recurrent 3/4)
//     preP         = bp + Wp[:, :128] @ ex_next
// K3: per-batch recurrent scan, h (256x128 f32) resident in LDS,
//     forget-gate GEMM h @ Wf_h^T done with V_WMMA_F32_16X16X4_F32.
// ---------------------------------------------------------------------------

typedef __attribute__((ext_vector_type(2))) float v2f;
typedef __attribute__((ext_vector_type(8))) float v8f;

#define B_   32
#define S_   128
#define DK   128
#define SK   256
#define HSTRIDE 132   // 128 + 4 pad: conflict-free ds_load_b64 fragments

#if defined(__has_builtin)
#if __has_builtin(__builtin_amdgcn_wmma_f32_16x16x4_f32)
#define HAVE_WMMA_F32 1
#endif
#endif

__device__ __forceinline__ float sigm_(float x) { return 1.0f / (1.0f + __expf(-x)); }

// ------------------------- K1: learning embedding --------------------------
__global__ __launch_bounds__(128) void lpkt_le_kernel(
    const int* __restrict__ exercise_id, const int* __restrict__ ans_time,
    const int* __restrict__ answer_value,
    const float* __restrict__ ex_table, const float* __restrict__ at_table,
    const float* __restrict__ W1, const float* __restrict__ b1,
    float* __restrict__ le)
{
  __shared__ float exr[DK], atr[DK];
  const int bs  = blockIdx.x;        // b*S + s
  const int tid = threadIdx.x;
  const int e  = exercise_id[bs];
  const int at = ans_time[bs];
  const float av = (float)answer_value[bs];
  exr[tid] = ex_table[e * DK + tid];
  atr[tid] = at_table[at * DK + tid];
  __syncthreads();
  const int o = tid;
  float acc = b1[o];
  float rs  = 0.f;
  const float4* w0 = (const float4*)(W1 + o * 384);
  const float4* w1 = (const float4*)(W1 + o * 384 + 128);
  const float4* w2 = (const float4*)(W1 + o * 384 + 256);
#pragma unroll 4
  for (int k4 = 0; k4 < 32; ++k4) {
    float4 a = w0[k4], b = w1[k4], c = w2[k4];
    const float* e4 = &exr[k4 * 4];
    const float* a4 = &atr[k4 * 4];
    acc += a.x * e4[0] + a.y * e4[1] + a.z * e4[2] + a.w * e4[3];
    acc += b.x * a4[0] + b.y * a4[1] + b.z * a4[2] + b.w * a4[3];
    rs  += c.x + c.y + c.z + c.w;    // ans is a broadcast scalar
  }
  le[bs * DK + o] = acc + av * rs;
}

// ---------------- K2: non-recurrent gate / predictor parts -----------------
__global__ __launch_bounds__(128) void lpkt_pre_kernel(
    const int* __restrict__ exercise_id, const int* __restrict__ interval_time,
    const float* __restrict__ ex_table, const float* __restrict__ it_table,
    const float* __restrict__ le,
    const float* __restrict__ Wl, const float* __restrict__ bl,
    const float* __restrict__ Wg, const float* __restrict__ bg,
    const float* __restrict__ Wp, const float* __restrict__ bp,
    float* __restrict__ preL, float* __restrict__ preG, float* __restrict__ preP)
{
  __shared__ float lp[DK], itv[DK], lv[DK], exn[DK];
  const int idx = blockIdx.x;        // b*127 + t
  const int b = idx / 127, t = idx % 127;
  const int tid = threadIdx.x;
  lp[tid]  = (t == 0) ? 0.f : le[(b * S_ + t - 1) * DK + tid];
  lv[tid]  = le[(b * S_ + t) * DK + tid];
  itv[tid] = it_table[interval_time[b * S_ + t + 1] * DK + tid];
  exn[tid] = ex_table[exercise_id[b * S_ + t + 1] * DK + tid];
  __syncthreads();
  const int o = tid;
  float accL = bl[o], accG = bg[o], accP = bp[o];
  const float4* wl0 = (const float4*)(Wl + o * 512);
  const float4* wl1 = (const float4*)(Wl + o * 512 + 128);
  const float4* wl2 = (const float4*)(Wl + o * 512 + 256);
  const float4* wg0 = (const float4*)(Wg + o * 512);
  const float4* wg1 = (const float4*)(Wg + o * 512 + 128);
  const float4* wg2 = (const float4*)(Wg + o * 512 + 256);
  const float4* wp0 = (const float4*)(Wp + o * 256);
#pragma unroll 2
  for (int k4 = 0; k4 < 32; ++k4) {
    const float* l4 = &lp[k4 * 4];
    const float* i4 = &itv[k4 * 4];
    const float* v4 = &lv[k4 * 4];
    const float* x4 = &exn[k4 * 4];
    float4 a;
    a = wl0[k4]; accL += a.x * l4[0] + a.y * l4[1] + a.z * l4[2] + a.w * l4[3];
    a = wl1[k4]; accL += a.x * i4[0] + a.y * i4[1] + a.z * i4[2] + a.w * i4[3];
    a = wl2[k4]; accL += a.x * v4[0] + a.y * v4[1] + a.z * v4[2] + a.w * v4[3];
    a = wg0[k4]; accG += a.x * l4[0] + a.y * l4[1] + a.z * l4[2] + a.w * l4[3];
    a = wg1[k4]; accG += a.x * i4[0] + a.y * i4[1] + a.z * i4[2] + a.w * i4[3];
    a = wg2[k4]; accG += a.x * v4[0] + a.y * v4[1] + a.z * v4[2] + a.w * v4[3];
    a = wp0[k4]; accP += a.x * x4[0] + a.y * x4[1] + a.z * x4[2] + a.w * x4[3];
  }
  preL[idx * DK + o] = accL;
  preG[idx * DK + o] = accG;
  preP[idx * DK + o] = accP;
}

// --------------------- K3: recurrent scan (1 WG / batch) -------------------
__global__ __launch_bounds__(512, 1) void lpkt_scan_kernel(
    const int* __restrict__ exercise_id, const int* __restrict__ interval_time,
    const float* __restrict__ q_matrix, const float* __restrict__ it_table,
    const float* __restrict__ Wl, const float* __restrict__ Wg,
    const float* __restrict__ Wf, const float* __restrict__ bf,
    const float* __restrict__ Wp, const float* __restrict__ h0,
    const float* __restrict__ preL, const float* __restrict__ preG,
    const float* __restrict__ preP, float* __restrict__ out)
{
  // 209 KB static LDS: legal only on CDNA5 (320 KB / WGP)
  __shared__ __align__(16) float sh_h[SK * HSTRIDE];   // 132 KB recurrent state
  __shared__ __align__(16) float sh_w[DK * HSTRIDE];   //  66 KB Wf_h (as [n][k])
  __shared__ float sh_kvt[SK], sh_kvn[SK], sh_itv[DK];
  __shared__ float sh_LG[DK], sh_fitv[DK], sh_fconst[DK], sh_ht[DK];
  __shared__ float sh_htp[4][DK];

  const int b = blockIdx.x;
  const int tid  = threadIdx.x;
  const int lane = tid & 31;
  const int wv   = tid >> 5;          // 16 waves
  const int lo   = lane & 15;
  const int hi   = lane >> 4;
  const int R    = wv * 16;           // each wave owns h rows [R, R+16)

  // ---- init: h0 -> LDS, Wf_h -> LDS, kv[:,0] -> kvn ----
  for (int i = tid; i < SK * DK; i += 512) {
    int m = i >> 7, k = i & 127;
    sh_h[m * HSTRIDE + k] = h0[i];
  }
  for (int i = tid; i < DK * DK; i += 512) {
    int o = i >> 7, k = i & 127;
    sh_w[o * HSTRIDE + k] = Wf[o * 384 + k];   // B[k][n] = Wf[n][k]
  }
  {
    int e0 = exercise_id[b * S_];
    for (int i = tid; i < SK; i += 512) sh_kvn[i] = q_matrix[e0 * SK + i];
  }
  __syncthreads();

  // ---- initial h_tilde = kv0 . h0 ----
  {
    int d = tid & 127, p = tid >> 7;
    float s = 0.f;
    for (int sk = p * 64; sk < p * 64 + 64; ++sk) s += sh_kvn[sk] * sh_h[sk * HSTRIDE + d];
    sh_htp[p][d] = s;
    __syncthreads();
    if (tid < 128) sh_ht[tid] = sh_htp[0][tid] + sh_htp[1][tid] + sh_htp[2][tid] + sh_htp[3][tid];
    __syncthreads();
  }

  for (int t = 0; t < 127; ++t) {
    const int pbase = (b * 127 + t) * DK;

    // ---- S0: stage per-step vectors ----
    const int e_t = exercise_id[b * S_ + t];
    const int e_n = exercise_id[b * S_ + t + 1];
    const int iid = interval_time[b * S_ + t + 1];
    if (tid < 256) sh_kvt[tid] = q_matrix[e_t * SK + tid];
    else           sh_kvn[tid - 256] = q_matrix[e_n * SK + (tid - 256)];
    if (tid < 128) sh_itv[tid] = it_table[iid * DK + tid];
    __syncthreads();

    // ---- A1: LG (recurrent quarter of the gate matvecs) + Wf_it part ----
    if (tid < 128) {
      const int o = tid;
      float accL = preL[pbase + o];
      float accG = preG[pbase + o];
      const float4* wl = (const float4*)(Wl + o * 512 + 384);
      const float4* wg = (const float4*)(Wg + o * 512 + 384);
#pragma unroll 4
      for (int k4 = 0; k4 < 32; ++k4) {
        float4 a = wl[k4], g = wg[k4];
        const float* hh = &sh_ht[k4 * 4];
        accL += a.x * hh[0] + a.y * hh[1] + a.z * hh[2] + a.w * hh[3];
        accG += g.x * hh[0] + g.y * hh[1] + g.z * hh[2] + g.w * hh[3];
      }
      float lg = tanhf(accL);
      float gt = sigm_(accG);
      sh_LG[o] = gt * (lg + 1.f) * 0.5f;
    } else if (tid < 256) {
      const int o = tid - 128;
      float acc = bf[o];
      const float4* wf = (const float4*)(Wf + o * 384 + 256);
#pragma unroll 4
      for (int k4 = 0; k4 < 32; ++k4) {
        float4 a = wf[k4];
        const float* ii = &sh_itv[k4 * 4];
        acc += a.x * ii[0] + a.y * ii[1] + a.z * ii[2] + a.w * ii[3];
      }
      sh_fitv[o] = acc;
    }
    __syncthreads();

    // ---- A2: fconst = bf + Wf_it@itv + Wf_LG@LG ----
    if (tid < 128) {
      const int o = tid;
      float acc = sh_fitv[o];
      const float4* wf = (const float4*)(Wf + o * 384 + 128);
#pragma unroll 4
      for (int k4 = 0; k4 < 32; ++k4) {
        float4 a = wf[k4];
        const float* gg = &sh_LG[k4 * 4];
        acc += a.x * gg[0] + a.y * gg[1] + a.z * gg[2] + a.w * gg[3];
      }
      sh_fconst[o] = acc;
    }
    __syncthreads();

    // ---- B: F = h(256x128) @ Wf_h^T via f32 WMMA; fused state update ----
    // Wave wv owns rows [R, R+16): reads/writes only its own rows -> no
    // intra-phase barrier needed.
    v8f acc[8] = {};
    const float* hrow = &sh_h[(R + lo) * HSTRIDE];
#if defined(HAVE_WMMA_F32)
    for (int k0 = 0; k0 < DK; k0 += 4) {
      // A frag: A[m=R+lo][k0 + 2*hi + j]   (16x4 f32, 2 VGPRs)
      v2f a = *(const v2f*)(hrow + k0 + 2 * hi);
#pragma unroll
      for (int nt = 0; nt < 8; ++nt) {
        // B frag: B[k][n] = Wf[n][k], n = nt*16 + lo
        v2f bb = *(const v2f*)(&sh_w[(nt * 16 + lo) * HSTRIDE + k0 + 2 * hi]);
        acc[nt] = __builtin_amdgcn_wmma_f32_16x16x4_f32(
            false, a, false, bb, (short)0, acc[nt], false, false);
      }
    }
#else
    // scalar fallback (compile-safety only)
#pragma unroll
    for (int nt = 0; nt < 8; ++nt)
#pragma unroll
      for (int r = 0; r < 8; ++r) {
        int m = R + r + 8 * hi, n = nt * 16 + lo;
        float s = 0.f;
        for (int k = 0; k < DK; ++k) s += sh_h[m * HSTRIDE + k] * sh_w[n * HSTRIDE + k];
        acc[nt][r] = s;
      }
#endif
    // f = sigmoid(F + fconst);  h = kvt (x) LG + f * h   (in place, own rows)
#pragma unroll
    for (int nt = 0; nt < 8; ++nt) {
      const int n = nt * 16 + lo;
      const float lgn = sh_LG[n];
      const float fcn = sh_fconst[n];
#pragma unroll
      for (int r = 0; r < 8; ++r) {
        const int m = R + r + 8 * hi;          // C/D layout: m = r + 8*hi
        const float f = sigm_(acc[nt][r] + fcn);
        const float hold = sh_h[m * HSTRIDE + n];
        sh_h[m * HSTRIDE + n] = sh_kvt[m] * lgn + f * hold;
      }
    }
    __syncthreads();

    // ---- C: h_tilde = kv_next . h ----
    {
      int d = tid & 127, p = tid >> 7;
      float s = 0.f;
      for (int sk = p * 64; sk < p * 64 + 64; ++sk) s += sh_kvn[sk] * sh_h[sk * HSTRIDE + d];
      sh_htp[p][d] = s;
    }
    __syncthreads();
    if (tid < 128) sh_ht[tid] = sh_htp[0][tid] + sh_htp[1][tid] + sh_htp[2][tid] + sh_htp[3][tid];
    __syncthreads();

    // ---- D: pred = sigmoid(preP + Wp_h @ h_tilde) ----
    if (tid < 128) {
      const int o = tid;
      float acc2 = preP[pbase + o];
      const float4* wp = (const float4*)(Wp + o * 256 + 128);
#pragma unroll 4
      for (int k4 = 0; k4 < 32; ++k4) {
        float4 a = wp[k4];
        const float* hh = &sh_ht[k4 * 4];
        acc2 += a.x * hh[0] + a.y * hh[1] + a.z * hh[2] + a.w * hh[3];
      }
      out[pbase + o] = sigm_(acc2);
    }
    __syncthreads();
  }
}

// ------------------------------- launcher ----------------------------------
extern "C" void kernel_launch(void* const* d_in, const int* in_sizes, int n_in,
                              void* d_out, int out_size, void* d_ws, size_t ws_size,
                              hipStream_t stream)
{
  const int*   exercise_id   = (const int*)d_in[0];
  const int*   ans_time      = (const int*)d_in[1];
  const int*   interval_time = (const int*)d_in[2];
  const int*   answer_value  = (const int*)d_in[3];
  const float* q_matrix = (const float*)d_in[4];
  const float* ex_table = (const float*)d_in[5];
  const float* at_table = (const float*)d_in[6];
  const float* it_table = (const float*)d_in[7];
  const float* W1 = (const float*)d_in[8];
  const float* b1 = (const float*)d_in[9];
  const float* Wl = (const float*)d_in[10];
  const float* bl = (const float*)d_in[11];
  const float* Wg = (const float*)d_in[12];
  const float* bg = (const float*)d_in[13];
  const float* Wf = (const float*)d_in[14];
  const float* bf = (const float*)d_in[15];
  const float* Wp = (const float*)d_in[16];
  const float* bp = (const float*)d_in[17];
  const float* h0 = (const float*)d_in[18];
  float* out = (float*)d_out;

  float* ws   = (float*)d_ws;
  float* le   = ws;                        // B*S*DK
  float* preL = le   + B_ * S_ * DK;       // B*127*DK
  float* preG = preL + B_ * 127 * DK;
  float* preP = preG + B_ * 127 * DK;

  lpkt_le_kernel<<<B_ * S_, 128, 0, stream>>>(
      exercise_id, ans_time, answer_value, ex_table, at_table, W1, b1, le);

  lpkt_pre_kernel<<<B_ * 127, 128, 0, stream>>>(
      exercise_id, interval_time, ex_table, it_table, le,
      Wl, bl, Wg, bg, Wp, bp, preL, preG, preP);

  lpkt_scan_kernel<<<B_, 512, 0, stream>>>(
      exercise_id, interval_time, q_matrix, it_table,
      Wl, Wg, Wf, bf, Wp, h0, preL, preG, preP, out);
}